// XAJMulTD_6270652252540
// MI455X (gfx1250) — compile-verified
//
#include <hip/hip_runtime.h>

// XAJ hydrological model scan for MI455X (gfx1250), TDM-streamed.
//
//   x          : (730, 512, 3)  f32     P = [...,0], PET = [...,2]
//   parameters : (730, 512, 12, 16) f32
//   out        : (730, 512)     f32     mean over mu of (RS + QI + QG)
//
// One thread per (grid, mu): 8192 threads = 256 wave32 waves.
// Block = 64 threads = 4 grids; per step the block's parameter slab is a
// contiguous 3072B run, so the Tensor Data Mover streams K steps per chunk
// into double-buffered LDS (2-D tile: 768 elems x K steps, stride = one t).
// Wave 0 issues the DMA for chunk c+1, waits TENSORcnt<=2 (chunk c's pair done),
// barriers, and all waves compute chunk c from LDS.

#define PRECS  1e-6f
#define NSTEP  730
#define NGRID  512
#define MU     16
#define KCH    8                       // time steps per TDM chunk
#define GPB    4                       // grids per block (64 threads)
#define PELEM  (GPB * 12 * MU)         // 768 param elements per step per block
#define XELEM  (GPB * 3)               // 12 x-elements per step per block

typedef __attribute__((ext_vector_type(4))) unsigned int u32x4;
typedef __attribute__((ext_vector_type(4))) int          i32x4;
typedef __attribute__((ext_vector_type(8))) int          i32x8;

// Issue one TDM 2-D tile load: `steps` rows of `width` 4-byte elements,
// row stride `stride_el` elements, from `gaddr` into LDS byte offset `lds_off`.
__device__ __forceinline__ void tdm_load_2d(const void* gaddr, unsigned lds_off,
                                            unsigned width, unsigned steps,
                                            unsigned long long stride_el)
{
    const unsigned long long ga = (unsigned long long)gaddr;

    u32x4 g0;
    g0.x = 1u;                                     // count=1, load, not restore
    g0.y = lds_off;                                // LDS byte address
    g0.z = (unsigned)(ga & 0xFFFFFFFFu);           // global_addr[31:0]
    g0.w = (unsigned)((ga >> 32) & 0x01FFFFFFu)    // global_addr[56:32]
         | (2u << 30);                             // type = 2 ("image")

    i32x8 g1;
    g1[0] = (int)(2u << 16);                       // data_size = 2 (4 bytes)
    g1[1] = (int)((width & 0xFFFFu) << 16);        // tensor_dim0[15:0]
    g1[2] = (int)(((width >> 16) & 0xFFFFu)        // tensor_dim0[31:16]
                  | ((steps & 0xFFFFu) << 16));    // tensor_dim1[15:0]
    g1[3] = (int)(((steps >> 16) & 0xFFFFu)        // tensor_dim1[31:16]
                  | ((width & 0xFFFFu) << 16));    // tile_dim0
    g1[4] = (int)(steps & 0xFFFFu);                // tile_dim1 (tile_dim2 = 0)
    g1[5] = (int)(unsigned)(stride_el & 0xFFFFFFFFull);   // dim0_stride[31:0]
    g1[6] = (int)(unsigned)((stride_el >> 32) & 0xFFFFull); // dim0_stride[47:32]
    g1[7] = 0;                                     // dim1_stride (unused, 2-D)

    i32x4 g2 = {0, 0, 0, 0};
    i32x4 g3 = {0, 0, 0, 0};

#if defined(__clang_major__) && (__clang_major__ >= 23)
    i32x8 g4 = {0, 0, 0, 0, 0, 0, 0, 0};
    __builtin_amdgcn_tensor_load_to_lds(g0, g1, g2, g3, g4, 0);
#else
    __builtin_amdgcn_tensor_load_to_lds(g0, g1, g2, g3, 0);
#endif
}

__global__ __launch_bounds__(64)
void xaj_scan_tdm_kernel(const float* __restrict__ x,
                         const float* __restrict__ params,
                         float* __restrict__ out)
{
    extern __shared__ float sbuf[];
    // dynamic LDS starts right after static LDS
    const unsigned lds_dyn = __builtin_amdgcn_groupstaticsize();

    const int tx = threadIdx.x;
    const int m  = tx & (MU - 1);      // mu index
    const int gl = tx >> 4;            // local grid 0..3
    const int g  = blockIdx.x * GPB + gl;

    // LDS layout (floats): [2][KCH][PELEM] params, then [2][KCH][XELEM] x
    const unsigned xoff_f = 2u * KCH * PELEM;
    const float* sp_base = sbuf;
    const float* sx_base = sbuf + xoff_f;

    // global bases for this block
    const float* gpar = params + (size_t)blockIdx.x * GPB * (12 * MU);
    const float* gx   = x + (size_t)blockIdx.x * GPB * 3;
    const unsigned long long par_tstride = (unsigned long long)NGRID * 12 * MU;
    const unsigned long long x_tstride   = (unsigned long long)NGRID * 3;

    const int nch = (NSTEP + KCH - 1) / KCH;

    // carry state, init 0.001
    float WU = 0.001f, WL = 0.001f, WD = 0.001f, S = 0.001f;
    float FR0 = 0.001f, QI = 0.001f, QG = 0.001f;

    // ---- prologue: stream chunk 0 into buffer 0 (wave 0 only)
    if (tx < 32) {
        tdm_load_2d(gpar, lds_dyn, PELEM, (unsigned)KCH, par_tstride);
        tdm_load_2d(gx, lds_dyn + xoff_f * 4u, XELEM, (unsigned)KCH, x_tstride);
    }

    for (int c = 0; c < nch; ++c) {
        const int buf = c & 1;
        const int t0  = c * KCH;
        const int steps = (NSTEP - t0 < KCH) ? (NSTEP - t0) : KCH;

        // buffer (buf^1) is free: everyone finished chunk c-1
        __syncthreads();
        if (tx < 32) {
            if (c + 1 < nch) {
                const int t1 = (c + 1) * KCH;
                const unsigned s1 = (unsigned)((NSTEP - t1 < KCH) ? (NSTEP - t1) : KCH);
                const int nb = buf ^ 1;
                tdm_load_2d(gpar + (unsigned long long)t1 * par_tstride,
                            lds_dyn + (unsigned)nb * (KCH * PELEM) * 4u,
                            PELEM, s1, par_tstride);
                tdm_load_2d(gx + (unsigned long long)t1 * x_tstride,
                            lds_dyn + (xoff_f + (unsigned)nb * (KCH * XELEM)) * 4u,
                            XELEM, s1, x_tstride);
                __builtin_amdgcn_s_wait_tensorcnt(2);   // chunk c's pair done
            } else {
                __builtin_amdgcn_s_wait_tensorcnt(0);   // last chunk done
            }
        }
        __syncthreads();                                // chunk c visible

        const float* sp = sp_base + buf * (KCH * PELEM) + gl * (12 * MU) + m;
        const float* sx = sx_base + buf * (KCH * XELEM) + gl * 3;

        for (int s = 0; s < steps; ++s) {
            const int t = t0 + s;
            const float* pp = sp + s * PELEM;

            const float parKE  = 0.3f  + pp[0 * MU]  * (2.0f   - 0.3f);
            const float parB   = 0.0f  + pp[1 * MU]  * (5.0f   - 0.0f);
            const float parWUM = 0.01f + pp[2 * MU]  * (100.0f - 0.01f);
            const float parWLM = 0.01f + pp[3 * MU]  * (100.0f - 0.01f);
            const float parWM  = 0.01f + pp[4 * MU]  * (200.0f - 0.01f);
            const float parC   = 0.09f + pp[5 * MU]  * (1.0f   - 0.09f);
            const float parSM  = 0.01f + pp[6 * MU]  * (100.0f - 0.01f);
            const float parEX  = 0.0f  + pp[7 * MU]  * (10.0f  - 0.0f);
            const float parKIr = 0.01f + pp[8 * MU]  * (0.7f   - 0.01f);
            const float parKGr = 0.01f + pp[9 * MU]  * (0.7f   - 0.01f);
            const float parCI  = 0.0f  + pp[10 * MU] * (0.998f - 0.0f);
            const float parCG  = 0.0f  + pp[11 * MU] * (0.998f - 0.0f);

            const float P_t   = sx[s * XELEM + 0];
            const float PET_t = sx[s * XELEM + 2] * parKE;

            const float wdm   = fmaxf(parWM - parWUM - parWLM, 0.0f);
            const float sum_k = parKIr + parKGr;
            const float parKI = (sum_k < 1.0f) ? parKIr
                              : (1.0f - PRECS) * parKIr / (sum_k + PRECS);
            const float parKG = (sum_k < 1.0f) ? parKGr
                              : (1.0f - PRECS) * parKGr / (sum_k + PRECS);

            // evaporation
            const float EU = fminf(PET_t, WU + P_t);
            const float D  = fmaxf(PET_t - EU, 0.0f);
            const float wet_excess = fmaxf(WL - parC * parWLM, 0.0f);
            const float wet_ratio  = wet_excess / (wet_excess + 1.0f);
            const float EL_wet = D * WL / (parWLM + PRECS);
            const float EL_dry = fminf(parC * D, WL);
            float EL = fmaxf(wet_ratio * EL_wet + (1.0f - wet_ratio) * EL_dry, 0.0f);
            EL = fminf(EL, WL);
            const float D_res = fmaxf(D - EL, 0.0f);
            const float ED = fminf(parC * D_res, WD);
            const float E  = EU + EL + ED;
            const float PE = fmaxf(P_t - E, 0.0f);

            // runoff generation (saturation-excess)
            const float W      = WU + WL + WD;
            const float WMM    = parWM * (1.0f + parB);
            const float base_W = fmaxf(1.0f - W / (parWM + PRECS), PRECS);
            const float A      = WMM * (1.0f - powf(base_W, 1.0f / (1.0f + parB)));
            const float ratio  = fminf((PE + A) / (WMM + PRECS), 1.0f);
            const float base_r = fmaxf(1.0f - ratio, PRECS);
            float R_per = (PE + A < WMM)
                ? (PE - (parWM - W) + parWM * powf(base_r, 1.0f + parB))
                : (PE - (parWM - W));
            R_per = fmaxf(R_per, 0.0f);

            // soil-moisture bookkeeping
            const float dW = P_t - E - R_per;
            float dpos = fmaxf(dW, 0.0f);
            const float fill_u = fminf(dpos, fmaxf(parWUM - WU, 0.0f));
            WU += fill_u; dpos -= fill_u;
            const float fill_l = fminf(dpos, fmaxf(parWLM - WL, 0.0f));
            WL += fill_l; dpos -= fill_l;
            const float fill_d = fminf(dpos, fmaxf(wdm - WD, 0.0f));
            WD += fill_d;
            float dneg = fmaxf(-dW, 0.0f);
            const float sub_u = fminf(dneg, WU); WU -= sub_u; dneg -= sub_u;
            const float sub_l = fminf(dneg, WL); WL -= sub_l; dneg -= sub_l;
            const float sub_d = fminf(dneg, WD); WD -= sub_d;

            // runoff separation (free-water reservoir)
            float FR = (PE > 0.0f) ? (R_per / (PE + PRECS)) : FR0;
            FR = fminf(fmaxf(FR, 0.0f), 1.0f);
            const float S_eq    = fminf(S * FR0 / (FR + PRECS), parSM);
            const float SMM     = parSM * (1.0f + parEX);
            const float ratio_s = fmaxf(1.0f - S_eq / (parSM + PRECS), PRECS);
            const float AU      = SMM * (1.0f - powf(ratio_s, 1.0f / (1.0f + parEX)));
            const float ratio_pa = fminf((PE + AU) / (SMM + PRECS), 1.0f);
            const float base_pa  = fmaxf(1.0f - ratio_pa, PRECS);
            float RS = (PE + AU < SMM)
                ? (PE + S_eq - parSM + parSM * powf(base_pa, 1.0f + parEX)) * FR
                : (PE + S_eq - parSM) * FR;
            RS = fminf(fmaxf(RS, 0.0f), R_per);
            S = fminf(S_eq + (R_per - RS) / (FR + PRECS), parSM);
            const float RI = parKI * S * FR;
            const float RG = parKG * S * FR;
            S  = S * (1.0f - parKI - parKG);
            QI = parCI * QI + (1.0f - parCI) * RI;
            QG = parCG * QG + (1.0f - parCG) * RG;
            FR0 = FR;

            // mean over mu: exact 16-lane xor-shuffle tree (2 grids per wave32)
            float q = RS + QI + QG;
            q += __shfl_xor(q, 1, 32);
            q += __shfl_xor(q, 2, 32);
            q += __shfl_xor(q, 4, 32);
            q += __shfl_xor(q, 8, 32);
            if (m == 0) out[(size_t)t * NGRID + g] = q * (1.0f / 16.0f);
        }
    }
}

extern "C" void kernel_launch(void* const* d_in, const int* in_sizes, int n_in,
                              void* d_out, int out_size, void* d_ws, size_t ws_size,
                              hipStream_t stream) {
    const float* x      = (const float*)d_in[0];   // (730,512,3) f32
    const float* params = (const float*)d_in[1];   // (730,512,12,16) f32
    float* out = (float*)d_out;                    // (730,512) f32

    const size_t shmem = (size_t)(2 * KCH * PELEM + 2 * KCH * XELEM) * sizeof(float);
    dim3 block(64);                                // 2 waves, 4 grids per block
    dim3 grid(NGRID / GPB);                        // 128 workgroups
    xaj_scan_tdm_kernel<<<grid, block, shmem, stream>>>(x, params, out);
}